// FlexVQCLayer_77781857730748
// MI455X (gfx1250) — compile-verified
//
#include <hip/hip_runtime.h>

typedef __attribute__((ext_vector_type(2))) float v2f;
typedef __attribute__((ext_vector_type(8))) float v8f;

#define N_QUBITS 8
#define DIM 256
#define N_LAYERS 2
#define WAVES_PER_BLOCK 8

// Entry of Kron(U0..U3) where Uq = RX: U[a,b] = cos if a==b else -i*sin.
// d = r XOR k (4 bits, bit3 <-> first qubit of the group).
// value = (-i)^popcount(d) * prod_j (bit_j ? sin_j : cos_j)
__device__ __forceinline__ void kron_entry(int d, const float fc[4], const float fs[4],
                                           float& re, float& im) {
  float m = ((d & 8) ? fs[0] : fc[0]) * ((d & 4) ? fs[1] : fc[1]) *
            ((d & 2) ? fs[2] : fc[2]) * ((d & 1) ? fs[3] : fc[3]);
  int h = __popc(d & 15) & 3;
  re = (h == 0) ? m : ((h == 2) ? -m : 0.0f);
  im = (h == 1) ? -m : ((h == 3) ? m : 0.0f);
}

__global__ __launch_bounds__(256) void vqc_kernel(const float* __restrict__ inputs,
                                                  const float* __restrict__ weights,
                                                  float* __restrict__ out, int batch) {
  __shared__ unsigned short sigma[DIM];              // CNOT-chain read permutation
  __shared__ float sre[WAVES_PER_BLOCK][DIM];        // per-wave state (real)
  __shared__ float sim[WAVES_PER_BLOCK][DIM];        // per-wave state (imag)

  const int tid = threadIdx.x;

  // ---- sigma: psi_after[x] = psi_before[sigma[x]] for CNOT(0,1)..CNOT(6,7) ----
  // qubit q <-> bit (7-q).  Innermost gate of the composed read-map is the LAST
  // circuit gate, so apply CNOT(6,7) first, CNOT(0,1) last.
  {
    int y = tid;
    #pragma unroll
    for (int i = 6; i >= 0; --i) {
      int cb = 1 << (7 - i);
      int tb = 1 << (6 - i);
      if (y & cb) y ^= tb;
    }
    sigma[tid] = (unsigned short)y;
  }
  __syncthreads();

  const int wave = tid >> 5;
  const int lane = tid & 31;
  const int b = blockIdx.x * WAVES_PER_BLOCK + wave;
  if (b >= batch) return;                 // wave-uniform: EXEC stays all-ones

  float* sr = sre[wave];
  float* si = sim[wave];

  const int tHi = lane >> 4;              // lane half (0/1)
  const int tLo = lane & 15;

  // ---- initial state: RY product state, S = u * v^T (real) ----
  {
    float ic[8], is_[8];
    #pragma unroll
    for (int q = 0; q < 8; ++q) {
      float t = 0.5f * inputs[b * 8 + q];
      ic[q] = __cosf(t);
      is_[q] = __sinf(t);
    }
    int r = lane >> 1;                    // this lane fills row r, cols c0..c0+7
    float u = ((r & 8) ? is_[0] : ic[0]) * ((r & 4) ? is_[1] : ic[1]) *
              ((r & 2) ? is_[2] : ic[2]) * ((r & 1) ? is_[3] : ic[3]);
    int c0 = (lane & 1) * 8;
    #pragma unroll
    for (int e = 0; e < 8; ++e) {
      int c = c0 + e;
      float v = ((c & 8) ? is_[4] : ic[4]) * ((c & 4) ? is_[5] : ic[5]) *
                ((c & 2) ? is_[6] : ic[6]) * ((c & 1) ? is_[7] : ic[7]);
      sr[r * 16 + c] = u * v;
      si[r * 16 + c] = 0.0f;
    }
  }
  // per-wave LDS traffic is in-order; no barrier needed past this point

  #pragma unroll
  for (int l = 0; l < N_LAYERS; ++l) {
    float fcA[4], fsA[4], fcB[4], fsB[4];
    #pragma unroll
    for (int q = 0; q < 4; ++q) {
      float ta = 0.5f * weights[l * 8 + q];
      fcA[q] = __cosf(ta); fsA[q] = __sinf(ta);
      float tb = 0.5f * weights[l * 8 + 4 + q];
      fcB[q] = __cosf(tb); fsB[q] = __sinf(tb);
    }

    // ---- left multiply: D = KronA * T, where T[r,c] = S[sigma[r*16+c]] ----
    v8f dRe = {}; v8f dIm = {};
    #pragma unroll
    for (int k = 0; k < 4; ++k) {
      v2f aRe, aIm, naIm, bRe, bIm;
      #pragma unroll
      for (int v = 0; v < 2; ++v) {
        int kk = 4 * k + v + 2 * tHi;     // global K index for this VGPR/half
        float re, im;
        kron_entry(tLo ^ kk, fcA, fsA, re, im);     // A[M=tLo, K=kk]
        aRe[v] = re; aIm[v] = im; naIm[v] = -im;
        int addr = sigma[kk * 16 + tLo];            // T row kk, col tLo
        bRe[v] = sr[addr];
        bIm[v] = si[addr];
      }
      dRe = __builtin_amdgcn_wmma_f32_16x16x4_f32(false, aRe,  false, bRe, (short)0, dRe, false, false);
      dRe = __builtin_amdgcn_wmma_f32_16x16x4_f32(false, naIm, false, bIm, (short)0, dRe, false, false);
      dIm = __builtin_amdgcn_wmma_f32_16x16x4_f32(false, aRe,  false, bIm, (short)0, dIm, false, false);
      dIm = __builtin_amdgcn_wmma_f32_16x16x4_f32(false, aIm,  false, bRe, (short)0, dIm, false, false);
    }

    // ---- stash D row-major in LDS (C/D layout: VGPR j holds rows j, j+8) ----
    #pragma unroll
    for (int j = 0; j < 8; ++j) {
      int addr = (j + 8 * tHi) * 16 + tLo;
      sr[addr] = dRe[j];
      si[addr] = dIm[j];
    }

    // ---- right multiply: E = D * KronB (D re-read as A-operand = transpose) ----
    v8f eRe = {}; v8f eIm = {};
    #pragma unroll
    for (int k = 0; k < 4; ++k) {
      v2f aRe, aIm, bRe, bIm, nbIm;
      #pragma unroll
      for (int v = 0; v < 2; ++v) {
        int kk = 4 * k + v + 2 * tHi;
        int addr = tLo * 16 + kk;                   // D[M=tLo, K=kk]
        aRe[v] = sr[addr];
        aIm[v] = si[addr];
        float re, im;
        kron_entry(tLo ^ kk, fcB, fsB, re, im);     // KronB[kk, tLo] (symmetric)
        bRe[v] = re; bIm[v] = im; nbIm[v] = -im;
      }
      eRe = __builtin_amdgcn_wmma_f32_16x16x4_f32(false, aRe, false, bRe,  (short)0, eRe, false, false);
      eRe = __builtin_amdgcn_wmma_f32_16x16x4_f32(false, aIm, false, nbIm, (short)0, eRe, false, false);
      eIm = __builtin_amdgcn_wmma_f32_16x16x4_f32(false, aRe, false, bIm,  (short)0, eIm, false, false);
      eIm = __builtin_amdgcn_wmma_f32_16x16x4_f32(false, aIm, false, bRe,  (short)0, eIm, false, false);
    }

    // ---- write E back as the new state ----
    #pragma unroll
    for (int j = 0; j < 8; ++j) {
      int addr = (j + 8 * tHi) * 16 + tLo;
      sr[addr] = eRe[j];
      si[addr] = eIm[j];
    }
  }

  // ---- Pauli-Z expectations: Z_q = sum p(idx) * (1 - 2*bit_{7-q}(idx)) ----
  float zs[8] = {0.f, 0.f, 0.f, 0.f, 0.f, 0.f, 0.f, 0.f};
  #pragma unroll
  for (int j = 0; j < 8; ++j) {
    int idx = (j + 8 * tHi) * 16 + tLo;
    float re = sr[idx], im = si[idx];
    float p = re * re + im * im;
    #pragma unroll
    for (int q = 0; q < 8; ++q) {
      float sgn = ((idx >> (7 - q)) & 1) ? -1.0f : 1.0f;
      zs[q] = __builtin_fmaf(sgn, p, zs[q]);
    }
  }
  #pragma unroll
  for (int q = 0; q < 8; ++q) {
    float v = zs[q];
    #pragma unroll
    for (int m = 16; m >= 1; m >>= 1) v += __shfl_xor(v, m, 32);
    if (lane == 0) out[b * 8 + q] = v;
  }
}

extern "C" void kernel_launch(void* const* d_in, const int* in_sizes, int n_in,
                              void* d_out, int out_size, void* d_ws, size_t ws_size,
                              hipStream_t stream) {
  (void)n_in; (void)out_size; (void)d_ws; (void)ws_size;
  const float* inputs = (const float*)d_in[0];   // (BATCH, 8) f32
  const float* weights = (const float*)d_in[1];  // (2, 8) f32
  float* out = (float*)d_out;                    // (BATCH, 8) f32
  int batch = in_sizes[0] / N_QUBITS;
  int blocks = (batch + WAVES_PER_BLOCK - 1) / WAVES_PER_BLOCK;
  hipLaunchKernelGGL(vqc_kernel, dim3(blocks), dim3(256), 0, stream,
                     inputs, weights, out, batch);
}